// EuclideanCodebook_10161892623007
// MI455X (gfx1250) — compile-verified
//
#include <hip/hip_runtime.h>
#include <stdint.h>

#define T_TOK   32768
#define D_DIM   128
#define K_CODES 8192
#define DECAYF  0.1f
#define EPSF    1e-5f

typedef __attribute__((ext_vector_type(2))) float v2f;
typedef __attribute__((ext_vector_type(8))) float v8f;

// ---------------------------------------------------------------- utilities
__device__ __forceinline__ unsigned long long pack_dist(float d, int col) {
    unsigned u = __float_as_uint(d);
    // monotone float -> uint map (handles negatives from rounding)
    u = (u & 0x80000000u) ? ~u : (u | 0x80000000u);
    return ((unsigned long long)u << 32) | (unsigned)col;
}

// ---------------------------------------------------------------- init scratch
__global__ void init_ws_kernel(unsigned long long* __restrict__ token_min,
                               float* __restrict__ cs, float* __restrict__ es,
                               float* __restrict__ sum) {
    size_t i = (size_t)blockIdx.x * blockDim.x + threadIdx.x;
    size_t stride = (size_t)gridDim.x * blockDim.x;
    for (size_t j = i; j < (size_t)K_CODES * D_DIM; j += stride) es[j] = 0.0f;
    for (size_t j = i; j < T_TOK; j += stride) token_min[j] = 0xFFFFFFFFFFFFFFFFull;
    for (size_t j = i; j < K_CODES; j += stride) cs[j] = 0.0f;
    if (i == 0) *sum = 0.0f;
}

// ---------------------------------------------------------------- row |v|^2
// one wave32 per row of 128 floats (float4 per lane)
__global__ void row_sumsq_kernel(const float* __restrict__ src,
                                 float* __restrict__ dst, int nrows) {
    int wave = (int)((blockIdx.x * blockDim.x + threadIdx.x) >> 5);
    int lane = threadIdx.x & 31;
    if (wave >= nrows) return;
    float4 v = ((const float4*)(src + (size_t)wave * D_DIM))[lane];
    float s = v.x * v.x + v.y * v.y + v.z * v.z + v.w * v.w;
    for (int m = 16; m; m >>= 1) s += __shfl_xor(s, m, 32);
    if (lane == 0) dst[wave] = s;
}

// ---------------------------------------------------------------- fused dist GEMM + argmin
// dist = |x|^2 + |e|^2 - 2 x.e   via  C0 = x2+e2,  B_lds = -2*e,  D = A*B + C
// grid = (128 N-strips, 256 M-groups), block = 256 (8 waves, one 16x64 strip each)
__launch_bounds__(256)
__global__ void dist_gemm_kernel(const float* __restrict__ x,
                                 const float* __restrict__ e,
                                 const float* __restrict__ x2,
                                 const float* __restrict__ e2,
                                 float* __restrict__ dist,
                                 unsigned long long* __restrict__ token_min) {
    __shared__ float eLDS[64 * D_DIM];          // 32 KB, B strip pre-scaled by -2

    const int lane   = threadIdx.x & 31;
    const int wave   = threadIdx.x >> 5;
    const int strip  = blockIdx.x;              // 0..127  (64 codes each)
    const int tile_m = blockIdx.y * 8 + wave;   // 0..2047 (16 tokens each)
    const int col_base = strip * 64;
    const int row_base = tile_m * 16;

    // stage contiguous 64x128 strip of embed into LDS, scaled by -2
    {
        const float4* src = (const float4*)(e + (size_t)col_base * D_DIM);
        float4* dstl = (float4*)eLDS;
        for (int i = threadIdx.x; i < 64 * D_DIM / 4; i += 256) {
            float4 v = src[i];
            v.x *= -2.0f; v.y *= -2.0f; v.z *= -2.0f; v.w *= -2.0f;
            dstl[i] = v;
        }
    }
    __syncthreads();

    const int myc   = lane & 15;                // col within 16-tile / row within A tile
    const int khalf = (lane >> 4) * 2;          // K sub-pair per ISA A/B layout
    const int rhalf = (lane >> 4) * 8;          // C/D row half

    // accumulator init: x2[row] + e2[col]
    const float e2v0 = e2[col_base +  0 + myc];
    const float e2v1 = e2[col_base + 16 + myc];
    const float e2v2 = e2[col_base + 32 + myc];
    const float e2v3 = e2[col_base + 48 + myc];
    v8f acc0, acc1, acc2, acc3;
    for (int r = 0; r < 8; ++r) {
        float xv = x2[row_base + rhalf + r];
        acc0[r] = xv + e2v0; acc1[r] = xv + e2v1;
        acc2[r] = xv + e2v2; acc3[r] = xv + e2v3;
    }

    // K loop: 32 steps of 4, A from global (b64/lane), B from LDS (b64/lane)
    const float* arow = x + (size_t)(row_base + myc) * D_DIM + khalf;
    const float* bcol = eLDS + khalf;
    for (int s = 0; s < 32; ++s) {
        v2f a  = *(const v2f*)(arow + 4 * s);
        const float* bs = bcol + 4 * s;
        v2f b0 = *(const v2f*)(bs + ( 0 + myc) * D_DIM);
        v2f b1 = *(const v2f*)(bs + (16 + myc) * D_DIM);
        v2f b2 = *(const v2f*)(bs + (32 + myc) * D_DIM);
        v2f b3 = *(const v2f*)(bs + (48 + myc) * D_DIM);
        acc0 = __builtin_amdgcn_wmma_f32_16x16x4_f32(false, a, false, b0, (short)0, acc0, false, false);
        acc1 = __builtin_amdgcn_wmma_f32_16x16x4_f32(false, a, false, b1, (short)0, acc1, false, false);
        acc2 = __builtin_amdgcn_wmma_f32_16x16x4_f32(false, a, false, b2, (short)0, acc2, false, false);
        acc3 = __builtin_amdgcn_wmma_f32_16x16x4_f32(false, a, false, b3, (short)0, acc3, false, false);
    }

    // streaming store of dist + wave-local argmin + one u64 atomic-min per row
    const size_t drow0 = (size_t)(row_base + rhalf) * K_CODES + col_base + myc;
    for (int r = 0; r < 8; ++r) {
        size_t off = drow0 + (size_t)r * K_CODES;
        __builtin_nontemporal_store(acc0[r], dist + off +  0);
        __builtin_nontemporal_store(acc1[r], dist + off + 16);
        __builtin_nontemporal_store(acc2[r], dist + off + 32);
        __builtin_nontemporal_store(acc3[r], dist + off + 48);

        unsigned long long p = pack_dist(acc0[r], col_base +  0 + myc);
        unsigned long long q = pack_dist(acc1[r], col_base + 16 + myc);
        if (q < p) p = q;
        q = pack_dist(acc2[r], col_base + 32 + myc); if (q < p) p = q;
        q = pack_dist(acc3[r], col_base + 48 + myc); if (q < p) p = q;
        // reduce across the 16-lane half (xor < 16 stays within half)
        for (int m = 1; m < 16; m <<= 1) {
            unsigned long long o = __shfl_xor(p, m, 32);
            if (o < p) p = o;
        }
        if (myc == 0)
            atomicMin(&token_min[row_base + rhalf + r], p);
    }
}

// ---------------------------------------------------------------- gather + segment sums
// one wave32 per token
__launch_bounds__(256)
__global__ void finalize_kernel(const float* __restrict__ x,
                                const float* __restrict__ e,
                                const unsigned long long* __restrict__ token_min,
                                float* __restrict__ quant_out,
                                float* __restrict__ idx_out,
                                float* __restrict__ cs, float* __restrict__ es) {
    int tok  = (int)((blockIdx.x * blockDim.x + threadIdx.x) >> 5);
    int lane = threadIdx.x & 31;
    if (tok >= T_TOK) return;
    unsigned idx = (unsigned)(token_min[tok] & 0xFFFFFFFFull);
    if (lane == 0) {
        idx_out[tok] = (float)idx;
        atomicAdd(cs + idx, 1.0f);
    }
    float4 ev = ((const float4*)(e + (size_t)idx * D_DIM))[lane];
    ((float4*)(quant_out + (size_t)tok * D_DIM))[lane] = ev;
    float4 xv = ((const float4*)(x + (size_t)tok * D_DIM))[lane];
    float* ep = es + (size_t)idx * D_DIM + lane * 4;
    atomicAdd(ep + 0, xv.x); atomicAdd(ep + 1, xv.y);
    atomicAdd(ep + 2, xv.z); atomicAdd(ep + 3, xv.w);
}

// ---------------------------------------------------------------- EMA cluster size (+ total sum)
__global__ void ema_cs_kernel(const float* __restrict__ cluster_size,
                              const float* __restrict__ cs,
                              float* __restrict__ ncs_out, float* __restrict__ sum) {
    int k = blockIdx.x * blockDim.x + threadIdx.x;
    if (k >= K_CODES) return;
    float v = DECAYF * cluster_size[k] + (1.0f - DECAYF) * cs[k];
    ncs_out[k] = v;
    float s = v;
    for (int m = 16; m; m >>= 1) s += __shfl_xor(s, m, 32);
    if ((threadIdx.x & 31) == 0) atomicAdd(sum, s);
}

// ---------------------------------------------------------------- EMA embed avg + laplace-normalized embed
__global__ void ema_embed_kernel(const float* __restrict__ embed_avg,
                                 const float* __restrict__ es,
                                 const float* __restrict__ ncs,
                                 const float* __restrict__ sum,
                                 float* __restrict__ nea_out,
                                 float* __restrict__ nemb_out) {
    size_t i = (size_t)blockIdx.x * blockDim.x + threadIdx.x;
    if (i >= (size_t)K_CODES * D_DIM) return;
    int k = (int)(i >> 7);
    float S = *sum;
    float norm = (ncs[k] + EPSF) / (S + (float)K_CODES * EPSF);
    float nea = DECAYF * embed_avg[i] + (1.0f - DECAYF) * es[i];
    nea_out[i] = nea;
    nemb_out[i] = nea / norm;
}

// ---------------------------------------------------------------- launcher
extern "C" void kernel_launch(void* const* d_in, const int* in_sizes, int n_in,
                              void* d_out, int out_size, void* d_ws, size_t ws_size,
                              hipStream_t stream) {
    const float* x            = (const float*)d_in[0]; // (8,4096,128)
    const float* embed        = (const float*)d_in[1]; // (1,8192,128)
    const float* cluster_size = (const float*)d_in[2]; // (1,8192)
    const float* embed_avg    = (const float*)d_in[3]; // (1,8192,128)

    float* out = (float*)d_out;
    // output layout (flat, return order)
    float* quant_out = out;                                           // T*D
    float* idx_out   = out + (size_t)T_TOK * D_DIM;                   // T
    float* dist_out  = idx_out + T_TOK;                               // T*K
    float* nemb_out  = dist_out + (size_t)T_TOK * K_CODES;            // K*D
    float* ncs_out   = nemb_out + (size_t)K_CODES * D_DIM;            // K
    float* nea_out   = ncs_out + K_CODES;                             // K*D

    // workspace layout
    char* ws = (char*)d_ws;
    unsigned long long* token_min = (unsigned long long*)ws;          // T*8   = 256 KB
    float* x2  = (float*)(ws + 262144);                               // T*4   = 128 KB
    float* e2  = (float*)(ws + 262144 + 131072);                      // K*4   =  32 KB
    float* cs  = (float*)(ws + 262144 + 131072 + 32768);              // K*4   =  32 KB
    float* es  = (float*)(ws + 262144 + 131072 + 65536);              // K*D*4 =   4 MB
    float* sum = (float*)(ws + 262144 + 131072 + 65536 + 4194304);    // 4 B

    init_ws_kernel<<<1024, 256, 0, stream>>>(token_min, cs, es, sum);
    row_sumsq_kernel<<<(T_TOK * 32) / 256, 256, 0, stream>>>(x, x2, T_TOK);
    row_sumsq_kernel<<<(K_CODES * 32) / 256, 256, 0, stream>>>(embed, e2, K_CODES);

    dim3 ggrid(K_CODES / 64, T_TOK / (16 * 8));   // (128, 256)
    dist_gemm_kernel<<<ggrid, 256, 0, stream>>>(x, embed, x2, e2, dist_out, token_min);

    finalize_kernel<<<(T_TOK * 32) / 256, 256, 0, stream>>>(x, embed, token_min,
                                                            quant_out, idx_out, cs, es);
    ema_cs_kernel<<<K_CODES / 256, 256, 0, stream>>>(cluster_size, cs, ncs_out, sum);
    ema_embed_kernel<<<(K_CODES * D_DIM) / 256, 256, 0, stream>>>(embed_avg, es, ncs_out,
                                                                  sum, nea_out, nemb_out);
}